// APPNPConv_26216480375301
// MI455X (gfx1250) — compile-verified
//
#include <hip/hip_runtime.h>
#include <math.h>

#define ALPHA 0.1f
#define KITER 10
#define DFEAT 32

typedef __attribute__((ext_vector_type(2))) float v2f;
typedef __attribute__((ext_vector_type(8))) float v8f;

// h <- feat, acc <- 0, deg/norm buffer <- 0
__global__ void appnp_init(const float* __restrict__ feat, float* __restrict__ h,
                           float* __restrict__ acc, float* __restrict__ norm, int N) {
    int i = blockIdx.x * blockDim.x + threadIdx.x;
    int total = N * DFEAT;
    if (i < total) { h[i] = feat[i]; acc[i] = 0.0f; }
    if (i < N) norm[i] = 0.0f;
}

// deg[dst[e]] += 1
__global__ void appnp_degree(const int* __restrict__ dst, float* __restrict__ deg, int E) {
    int i = blockIdx.x * blockDim.x + threadIdx.x;
    if (i < E) atomicAdd(&deg[dst[i]], 1.0f);
}

// norm = deg > 0 ? deg^-1/2 : 0   (in place)
__global__ void appnp_norm(float* __restrict__ norm, int N) {
    int i = blockIdx.x * blockDim.x + threadIdx.x;
    if (i < N) {
        float d = norm[i];
        norm[i] = (d > 0.0f) ? (1.0f / sqrtf(d)) : 0.0f;
    }
}

// One wave per edge, lane d = feature d: acc[dst] += h[src]*norm[src]  (128B coalesced)
__global__ void appnp_scatter(const float* __restrict__ h, const float* __restrict__ norm,
                              const int* __restrict__ src, const int* __restrict__ dst,
                              float* __restrict__ acc, int E) {
    int wave = blockIdx.x * (blockDim.x >> 5) + (threadIdx.x >> 5);
    if (wave >= E) return;
    int lane = threadIdx.x & 31;
    if (lane == 0) {
        // stream the edge index arrays ahead (speculative global_prefetch_b8)
        __builtin_prefetch(src + wave + 4096, 0, 0);
        __builtin_prefetch(dst + wave + 4096, 0, 0);
    }
    int s = src[wave];
    int d = dst[wave];
    float v = h[s * DFEAT + lane] * norm[s];
    atomicAdd(&acc[d * DFEAT + lane], v);  // global_atomic_add_f32, no return
}

// One wave per 16-node tile (16 nodes x 32 feats = two 16x16 f32 tiles):
//   h = (1-a)*diag(norm)*acc + a*feat     via 4x V_WMMA_F32_16X16X4_F32 per tile,
// and re-zero acc for the next iteration.
__global__ void appnp_update_wmma(const float* __restrict__ feat, const float* __restrict__ norm,
                                  float* __restrict__ acc, float* __restrict__ h,
                                  int N, int ntiles) {
    int wave = blockIdx.x * (blockDim.x >> 5) + (threadIdx.x >> 5);
    if (wave >= ntiles) return;              // wave-uniform: EXEC all-ones below
    int node0 = wave * 16;
    int lane = threadIdx.x & 31;
    int hi = lane >> 4;                      // lane half
    int lo = lane & 15;

    if (node0 + 16 <= N) {
        float nv = (1.0f - ALPHA) * norm[node0 + lo];   // row scale for M = lo
#pragma unroll
        for (int fh = 0; fh < 2; ++fh) {
            int f0 = fh * 16;
            // C = alpha * feat tile  (C/D layout: VGPR r, lane l -> M=r+8*hi, N=lo)
            v8f cacc;
#pragma unroll
            for (int r = 0; r < 8; ++r)
                cacc[r] = ALPHA * feat[(node0 + r + 8 * hi) * DFEAT + f0 + lo];
            // D = sum_c A_c * B_c + C, A_c = 16x4 slice of diag((1-a)norm),
            // B_c = rows [4c..4c+3] of acc tile
#pragma unroll
            for (int cc = 0; cc < 4; ++cc) {
                v2f a, b;
#pragma unroll
                for (int v = 0; v < 2; ++v) {
                    int k = v + 2 * hi;                       // A/B K index per ISA layout
                    int m = 4 * cc + k;                       // global tile row
                    b[v] = acc[(node0 + m) * DFEAT + f0 + lo];
                    a[v] = (lo == (4 * cc + k)) ? nv : 0.0f;  // diagonal slice
                }
                cacc = __builtin_amdgcn_wmma_f32_16x16x4_f32(
                    /*neg_a=*/false, a, /*neg_b=*/false, b,
                    /*c_mod=*/(short)0, cacc, /*reuse_a=*/false, /*reuse_b=*/false);
            }
#pragma unroll
            for (int r = 0; r < 8; ++r)
                h[(node0 + r + 8 * hi) * DFEAT + f0 + lo] = cacc[r];
            // zero acc tile for next iteration (each element covered exactly once)
#pragma unroll
            for (int cc = 0; cc < 4; ++cc)
#pragma unroll
                for (int v = 0; v < 2; ++v) {
                    int m = 4 * cc + v + 2 * hi;
                    acc[(node0 + m) * DFEAT + f0 + lo] = 0.0f;
                }
        }
    } else {
        // scalar tail tile (lane = feature)
        for (int m = 0; m < 16; ++m) {
            int node = node0 + m;
            if (node < N) {
                int j = node * DFEAT + lane;
                h[j] = (1.0f - ALPHA) * norm[node] * acc[j] + ALPHA * feat[j];
                acc[j] = 0.0f;
            }
        }
    }
}

extern "C" void kernel_launch(void* const* d_in, const int* in_sizes, int n_in,
                              void* d_out, int out_size, void* d_ws, size_t ws_size,
                              hipStream_t stream) {
    const float* feat = (const float*)d_in[0];
    const int*   src  = (const int*)d_in[1];
    const int*   dst  = (const int*)d_in[2];
    int N = in_sizes[0] / DFEAT;
    int E = in_sizes[1];

    float* h    = (float*)d_out;                    // h lives in d_out
    float* acc  = (float*)d_ws;                     // N*32 f32 accumulator
    float* norm = acc + (size_t)N * DFEAT;          // N f32 (deg, then deg^-1/2)

    int totalND = N * DFEAT;
    appnp_init<<<(totalND + 255) / 256, 256, 0, stream>>>(feat, h, acc, norm, N);
    appnp_degree<<<(E + 255) / 256, 256, 0, stream>>>(dst, norm, E);
    appnp_norm<<<(N + 255) / 256, 256, 0, stream>>>(norm, N);

    int ntiles = (N + 15) / 16;
    for (int k = 0; k < KITER; ++k) {
        appnp_scatter<<<(E + 7) / 8, 256, 0, stream>>>(h, norm, src, dst, acc, E);
        appnp_update_wmma<<<(ntiles + 7) / 8, 256, 0, stream>>>(feat, norm, acc, h, N, ntiles);
    }
}